// Tatt_66778151518718
// MI455X (gfx1250) — compile-verified
//
#include <hip/hip_runtime.h>

// ---------------------------------------------------------------------------
// CDNA5 (gfx1250) temporal-GCN + Conv1d. fp32 WMMA (v_wmma_f32_16x16x4_f32),
// fully branchless hot loops via padded/transposed operands.
// ---------------------------------------------------------------------------

typedef __attribute__((ext_vector_type(2))) float v2f;
typedef __attribute__((ext_vector_type(8))) float v8f;

#define TIMELEN 2016
#define T1      288
#define T3      864
#define NCH     325                  // true channel count
#define NCHP    328                  // K padded to multiple of 4 (row stride)
#define NTP     336                  // N padded to multiple of 16 (21 tiles)
#define BATCH   32
#define KSIZE   577
#define PCONV   288
#define MTOT    (BATCH * T3)         // 27648
#define SZP     (MTOT * NCHP)        // padded feature buffer elements
#define SZS     (MTOT * NTP)         // padded support buffer elements
#define WTSZ    (NTP * NCHP)         // padded transposed weight elements

__device__ __forceinline__ float deg_of(int i) {
    int lo = i < 3 ? i : 3;
    int hi = (T3 - 1 - i) < 3 ? (T3 - 1 - i) : 3;
    return (float)(lo + hi + 1);
}
__device__ __forceinline__ int imin(int a, int b) { return a < b ? a : b; }

__device__ __forceinline__ v8f wmma4(v2f a, v2f b, v8f c) {
    return __builtin_amdgcn_wmma_f32_16x16x4_f32(false, a, false, b,
                                                 (short)0, c, false, false);
}

// ---------------------------------------------------------------------------
// Weight pad+transpose: Wt[n][k] = W[k][n], zero-filled to [336][328].
// ---------------------------------------------------------------------------
__global__ void pad_transpose_w_kernel(const float* __restrict__ W,
                                       float* __restrict__ Wt)
{
    int idx = blockIdx.x * blockDim.x + threadIdx.x;
    if (idx >= WTSZ) return;
    int k = idx % NCHP;
    int n = idx / NCHP;
    Wt[idx] = (n < NCH && k < NCH) ? W[k * NCH + n] : 0.f;
}

// ---------------------------------------------------------------------------
// Stage 1: temp[b,m,d] gather into padded [B,864,328] layout.
// ---------------------------------------------------------------------------
__global__ void build_temp_kernel(const float* __restrict__ inp,   // [B,325,288]
                                  const float* __restrict__ x,     // [2016,325]
                                  const int*   __restrict__ week,
                                  const int*   __restrict__ hour,
                                  float* __restrict__ temp)        // [B,864,328]
{
    int idx = blockIdx.x * blockDim.x + threadIdx.x;
    if (idx >= SZP) return;
    int d = idx % NCHP;
    int m = (idx / NCHP) % T3;
    int b = idx / (NCHP * T3);
    float v = 0.f;
    if (d < NCH) {
        if (m >= T1 && m < 2 * T1) {
            v = inp[(b * NCH + d) * T1 + (m - T1)];
        } else {
            int s = week[b] * T1 + hour[b];
            int t = (m + s - T1) % TIMELEN;
            if (t < 0) t += TIMELEN;
            v = x[t * NCH + d];
        }
    }
    temp[idx] = v;
}

// ---------------------------------------------------------------------------
// Stage 2a: dual GEMM  S1 = X @ W1, S2 = X @ W2 using padded operands.
// X: [27648,328] padded features; Wt: [336][328] transposed padded weights.
// One wave per 16x16 tile; 82 branchless K-steps; shared A operand.
// ---------------------------------------------------------------------------
__global__ void gemm_dual_kernel(const float* __restrict__ X,
                                 const float* __restrict__ Wt1,
                                 const float* __restrict__ Wt2,
                                 float* __restrict__ S1,   // [27648,336]
                                 float* __restrict__ S2)
{
    const int lane = threadIdx.x & 31;
    const int half = lane >> 4;          // K sub-pair selector
    const int rc   = lane & 15;          // A row / B-D column
    const int m0   = blockIdx.x * 16;
    const int n0   = blockIdx.y * 16;
    const int n    = n0 + rc;

    v8f acc1 = {};
    v8f acc2 = {};
    const float* Xr  = X   + (size_t)(m0 + rc) * NCHP;
    const float* W1r = Wt1 + (size_t)n * NCHP;
    const float* W2r = Wt2 + (size_t)n * NCHP;

    for (int kk = 0; kk < NCHP; kk += 4) {
        const int k0 = kk + 2 * half;
        const v2f a  = *(const v2f*)(Xr  + k0);
        const v2f b1 = *(const v2f*)(W1r + k0);
        const v2f b2 = *(const v2f*)(W2r + k0);
        acc1 = wmma4(a, b1, acc1);
        acc2 = wmma4(a, b2, acc2);
    }

    // n < 336 always valid in padded S; pad columns hold exact zeros.
#pragma unroll
    for (int v = 0; v < 8; ++v) {
        const int om = m0 + v + 8 * half;
        S1[(size_t)om * NTP + n] = acc1[v];
        S2[(size_t)om * NTP + n] = acc2[v];
    }
}

// ---------------------------------------------------------------------------
// Stage 2b: banded A-multiply + bias + tanh*sigmoid gate (padded layout).
// ---------------------------------------------------------------------------
__global__ void band_gate_kernel(const float* __restrict__ S1,   // [B*864,336]
                                 const float* __restrict__ S2,
                                 const float* __restrict__ bias1,
                                 const float* __restrict__ bias2,
                                 float* __restrict__ out)        // [B,864,328]
{
    int idx = blockIdx.x * blockDim.x + threadIdx.x;
    if (idx >= SZP) return;
    int o = idx % NCHP;
    int m = (idx / NCHP) % T3;
    int b = idx / (NCHP * T3);

    if (o >= NCH) { out[idx] = 0.f; return; }

    const float rm = rsqrtf(deg_of(m));
    const int jlo = (m - 3 < 0) ? 0 : m - 3;
    const int jhi = (m + 3 > T3 - 1) ? T3 - 1 : m + 3;

    float g1 = 0.f, g2 = 0.f;
    for (int j = jlo; j <= jhi; ++j) {
        const float cf = rm * rsqrtf(deg_of(j));
        const size_t off = ((size_t)b * T3 + j) * NTP + o;
        g1 += cf * S1[off];
        g2 += cf * S2[off];
    }
    const float t  = tanhf(g1 + bias1[o]);
    const float sg = 1.f / (1.f + expf(-(g2 + bias2[o])));
    out[idx] = t * sg;
}

// ---------------------------------------------------------------------------
// Stage 3: Conv1d(325->325, k=577, valid) as implicit GEMM.
// Grid (o-tile, p-pair, batch); 4 waves split the k-loop. Each wave computes
// TWO p-tiles sharing the A fragment (halves scattered conv_w loads per WMMA)
// with 4 independent accumulator chains; 40 unclamped c-steps + 1 clamped
// tail; LDS reduction of the 4 K-partials.
// ---------------------------------------------------------------------------
__global__ void conv_kernel(const float* __restrict__ X3,    // [B,864,328]
                            const float* __restrict__ W,     // [325,325,577]
                            const float* __restrict__ cbias, // [325]
                            float* __restrict__ out)         // [B,325,288]
{
    __shared__ float red[4 * 512];

    const int tid  = threadIdx.x;
    const int lane = tid & 31;
    const int wid  = tid >> 5;
    const int half = lane >> 4;
    const int rc   = lane & 15;
    const int o0   = blockIdx.x * 16;
    const int p0   = blockIdx.y * 32;          // two adjacent 16-wide p tiles
    const int bb   = blockIdx.z;
    const int oc   = imin(o0 + rc, NCH - 1);   // clamped A row (unused rows)

    v8f acc00 = {}, acc01 = {}, acc10 = {}, acc11 = {};
    const float* Xb = X3 + (size_t)bb * T3 * NCHP;
    const float* Wo = W + ((size_t)oc * NCH + 2 * half) * KSIZE;

    for (int k = wid; k < KSIZE; k += 4) {
        const float* xr0 = Xb + (size_t)(p0 + rc + k) * NCHP + 2 * half;
        const float* xr1 = xr0 + 16 * NCHP;
        const float* wp  = Wo + k;

        // 40 branchless steps: c = 0..319
        for (int cb = 0; cb < 320; cb += 8) {
            v2f a0, a1;
            a0.x = wp[0];                 // c = cb + 2h
            a0.y = wp[KSIZE];             // c = cb + 2h + 1
            a1.x = wp[4 * KSIZE];         // c = cb + 4 + 2h
            a1.y = wp[5 * KSIZE];
            const v2f b00 = *(const v2f*)(xr0 + cb);
            const v2f b01 = *(const v2f*)(xr0 + cb + 4);
            const v2f b10 = *(const v2f*)(xr1 + cb);
            const v2f b11 = *(const v2f*)(xr1 + cb + 4);
            acc00 = wmma4(a0, b00, acc00);
            acc10 = wmma4(a0, b10, acc10);
            acc01 = wmma4(a1, b01, acc01);
            acc11 = wmma4(a1, b11, acc11);
            wp += 8 * KSIZE;
        }

        // Tail step (c = 320..327): clamped A rows multiply zero-padded B rows.
        {
            v2f a0, a1;
            a0.x = wp[0];                 // c = 320 + 2h   (<= 323, valid)
            a0.y = wp[KSIZE];             // c = 321 + 2h   (<= 324, valid)
            const float aw = wp[(size_t)(4 - 2 * half) * KSIZE];  // value W[o][324]
            a1.x = aw;                    // c = 324+2h (clamped; h=0 real c=324)
            a1.y = aw;                    // c = 325+2h (always x zero B row)
            const v2f b00 = *(const v2f*)(xr0 + 320);
            const v2f b01 = *(const v2f*)(xr0 + 324);   // pads are zeros
            const v2f b10 = *(const v2f*)(xr1 + 320);
            const v2f b11 = *(const v2f*)(xr1 + 324);
            acc00 = wmma4(a0, b00, acc00);
            acc10 = wmma4(a0, b10, acc10);
            acc01 = wmma4(a1, b01, acc01);
            acc11 = wmma4(a1, b11, acc11);
        }
    }
    const v8f accP0 = acc00 + acc01;
    const v8f accP1 = acc10 + acc11;

    // Cross-wave reduction of the 4 K-partials through LDS.
#pragma unroll
    for (int v = 0; v < 8; ++v) {
        red[wid * 512 +       v * 32 + lane] = accP0[v];
        red[wid * 512 + 256 + v * 32 + lane] = accP1[v];
    }
    __syncthreads();

    for (int e = tid; e < 512; e += 128) {
        float s = red[e] + red[512 + e] + red[1024 + e] + red[1536 + e];
        const int pt = e >> 8;
        const int v  = (e >> 5) & 7;
        const int l  = e & 31;
        const int oo = o0 + v + 8 * (l >> 4);
        const int pp = p0 + pt * 16 + (l & 15);
        if (oo < NCH)
            out[((size_t)bb * NCH + oo) * PCONV + pp] = s + cbias[oo];
    }
}

// ---------------------------------------------------------------------------
// Host-side launch
// ---------------------------------------------------------------------------
extern "C" void kernel_launch(void* const* d_in, const int* in_sizes, int n_in,
                              void* d_out, int out_size, void* d_ws, size_t ws_size,
                              hipStream_t stream)
{
    const float* inp    = (const float*)d_in[0];
    const float* x      = (const float*)d_in[1];
    const float* w1     = (const float*)d_in[2];
    const float* b1     = (const float*)d_in[3];
    const float* w11    = (const float*)d_in[4];
    const float* b11    = (const float*)d_in[5];
    const float* w2     = (const float*)d_in[6];
    const float* b2     = (const float*)d_in[7];
    const float* w22    = (const float*)d_in[8];
    const float* b22    = (const float*)d_in[9];
    const float* w3     = (const float*)d_in[10];
    const float* b3     = (const float*)d_in[11];
    const float* w33    = (const float*)d_in[12];
    const float* b33    = (const float*)d_in[13];
    const float* conv_w = (const float*)d_in[14];
    const float* conv_b = (const float*)d_in[15];
    const int*   week   = (const int*)d_in[16];
    const int*   hour   = (const int*)d_in[17];
    float* out = (float*)d_out;

    float* ws = (float*)d_ws;
    float* F0  = ws;                          // padded feature ping [27648,328]
    float* F1  = F0 + (size_t)SZP;            // padded feature pong
    float* S1  = F1 + (size_t)SZP;            // padded support 1 [27648,336]
    float* S2  = S1 + (size_t)SZS;            // padded support 2
    float* Wt[6];
    Wt[0] = S2 + (size_t)SZS;
    for (int i = 1; i < 6; ++i) Wt[i] = Wt[i - 1] + (size_t)WTSZ;
    const float* wsrc[6] = { w1, w11, w2, w22, w3, w33 };

    const int padBlocks  = (WTSZ + 255) / 256;
    const int elemBlocks = (SZP + 255) / 256;
    const dim3 gemmGrid(MTOT / 16, NTP / 16);          // 1728 x 21
    const dim3 convGrid(NTP / 16, PCONV / 32, BATCH);  // 21 x 9 x 32

    // Weight pad+transpose (tiny).
    for (int i = 0; i < 6; ++i)
        pad_transpose_w_kernel<<<padBlocks, 256, 0, stream>>>(wsrc[i], Wt[i]);

    // Stage 1: build temp (gather) into padded layout.
    build_temp_kernel<<<elemBlocks, 256, 0, stream>>>(inp, x, week, hour, F0);

    // Layer 1
    gemm_dual_kernel<<<gemmGrid, 32, 0, stream>>>(F0, Wt[0], Wt[1], S1, S2);
    band_gate_kernel<<<elemBlocks, 256, 0, stream>>>(S1, S2, b1, b11, F1);
    // Layer 2
    gemm_dual_kernel<<<gemmGrid, 32, 0, stream>>>(F1, Wt[2], Wt[3], S1, S2);
    band_gate_kernel<<<elemBlocks, 256, 0, stream>>>(S1, S2, b2, b22, F0);
    // Layer 3
    gemm_dual_kernel<<<gemmGrid, 32, 0, stream>>>(F0, Wt[4], Wt[5], S1, S2);
    band_gate_kernel<<<elemBlocks, 256, 0, stream>>>(S1, S2, b3, b33, F1);

    // Conv1d (dominant stage, ~1.12 TFLOP; floor = 244 MB conv_w / 23.3 TB/s).
    conv_kernel<<<convGrid, 128, 0, stream>>>(F1, conv_w, conv_b, out);
}